// DynamicGrainedEncoder_34840774705287
// MI455X (gfx1250) — compile-verified
//
#include <hip/hip_runtime.h>

typedef __attribute__((ext_vector_type(2))) float v2f;
typedef __attribute__((ext_vector_type(8))) float v8f;

#define BB 32
#define HH 56
#define WW 56
#define CC 384
#define C4 96            // CC / 4 (float4 units)
#define NN 3136          // HH*WW
#define MM 4116          // 196 + 784 + 3136
#define HR 14
#define WR 14
#define H2 28
#define W2 28
#define ROWS 6272        // BB*HR*WR
#define Y_OFF1 196       // token offset of s=2 grain
#define Y_OFF2 980       // token offset of full-res 56x56 grain

// ---------------- Kernel 1: 2x2 average pool of x -> pooled2 [B,28,28,C] ----
__global__ __launch_bounds__(256) void dge_pool2(const float* __restrict__ x,
                                                 float* __restrict__ p2) {
  int t = blockIdx.x * 256 + threadIdx.x;
  const int total = BB * H2 * W2 * C4;
  if (t >= total) return;
  int c4 = t % C4;
  int w2 = (t / C4) % W2;
  int h2 = (t / (C4 * W2)) % H2;
  int b  =  t / (C4 * W2 * H2);
  const float4* x4 = (const float4*)x;
  size_t base = (((size_t)b * NN) + (size_t)(2 * h2) * WW + (size_t)(2 * w2)) * C4 + c4;
  float4 a0 = x4[base];
  float4 a1 = x4[base + C4];
  float4 a2 = x4[base + (size_t)WW * C4];
  float4 a3 = x4[base + (size_t)(WW + 1) * C4];
  float4 r;
  r.x = (a0.x + a1.x + a2.x + a3.x) * 0.25f;
  r.y = (a0.y + a1.y + a2.y + a3.y) * 0.25f;
  r.z = (a0.z + a1.z + a2.z + a3.z) * 0.25f;
  r.w = (a0.w + a1.w + a2.w + a3.w) * 0.25f;
  ((float4*)p2)[t] = r;
}

// ---------------- Kernel 2: 2x2 pool of pooled2 -> pooled4 [B,14,14,C] ------
__global__ __launch_bounds__(256) void dge_pool4(const float* __restrict__ p2,
                                                 float* __restrict__ p4) {
  int t = blockIdx.x * 256 + threadIdx.x;
  const int total = BB * HR * WR * C4;
  if (t >= total) return;
  int c4 = t % C4;
  int wr = (t / C4) % WR;
  int hr = (t / (C4 * WR)) % HR;
  int b  =  t / (C4 * WR * HR);
  const float4* s4 = (const float4*)p2;
  size_t base = (((size_t)b * H2 * W2) + (size_t)(2 * hr) * W2 + (size_t)(2 * wr)) * C4 + c4;
  float4 a0 = s4[base];
  float4 a1 = s4[base + C4];
  float4 a2 = s4[base + (size_t)W2 * C4];
  float4 a3 = s4[base + (size_t)(W2 + 1) * C4];
  float4 r;
  r.x = (a0.x + a1.x + a2.x + a3.x) * 0.25f;
  r.y = (a0.y + a1.y + a2.y + a3.y) * 0.25f;
  r.z = (a0.z + a1.z + a2.z + a3.z) * 0.25f;
  r.w = (a0.w + a1.w + a2.w + a3.w) * 0.25f;
  ((float4*)p4)[t] = r;
}

// ---------------- Kernel 3: router logits via f32 WMMA + hard argmax --------
// Rows = 6272 pooled4 vectors (C=384); cols 0..2 = gate_w rows; K stepped by 4
// using V_WMMA_F32_16X16X4_F32 (exact f32 math; argmax is precision-sensitive).
// B (gate_w) is staged once into LDS padded to 16 rows (rows 3..15 = 0) so the
// inner loop is branch-free: global_load_b64 (A) + ds_load_b64 (B) + v_wmma.
__global__ __launch_bounds__(256) void dge_gate(const float* __restrict__ p4,
                                                const float* __restrict__ gw,
                                                const float* __restrict__ gb,
                                                int* __restrict__ sel) {
  __shared__ float bsh[16 * CC];       // 24 KB: gate_w padded to 16x384
  __shared__ float dsh[8][16][4];      // per-wave D spill for argmax

  // Cooperative fill of padded B (once per block).
  for (int i = threadIdx.x; i < 16 * CC; i += 256) {
    int row = i / CC;
    int col = i - row * CC;
    bsh[i] = (row < 3) ? gw[row * CC + col] : 0.0f;
  }
  __syncthreads();

  int lane = threadIdx.x & 31;
  int wave = threadIdx.x >> 5;
  int tile = blockIdx.x * 8 + wave;    // 392 tiles total
  int m0   = tile * 16;                // row base
  int r    = lane & 15;                // A: row index / B: col index
  int kb   = (lane >> 4) << 1;         // lanes 0-15 -> K=0,1 ; lanes 16-31 -> K=2,3

  const float* arow = p4  + (size_t)(m0 + r) * CC + kb;
  const float* brow = bsh + r * CC + kb;

  v8f acc = {};
#pragma unroll 4
  for (int k = 0; k < CC; k += 4) {
    v2f a  = *(const v2f*)(arow + k);  // 8B aligned: k%4==0, kb even
    v2f bm = *(const v2f*)(brow + k);  // LDS, branch-free
    // (neg_a, A, neg_b, B, c_mod, C, reuse_a, reuse_b)
    acc = __builtin_amdgcn_wmma_f32_16x16x4_f32(false, a, false, bm,
                                                (short)0, acc, false, false);
  }

  // D layout: VGPR j holds rows j (lanes 0-15) and j+8 (lanes 16-31), col = lane&15
  if (r < 3) {
    int mh = (lane >> 4) * 8;
#pragma unroll
    for (int j = 0; j < 8; ++j) dsh[wave][mh + j][r] = acc[j];
  }
  __syncthreads();
  if (lane < 16) {
    float l0 = dsh[wave][lane][0] + gb[0];
    float l1 = dsh[wave][lane][1] + gb[1];
    float l2 = dsh[wave][lane][2] + gb[2];
    int s = 0; float best = l0;        // first-max tie-break (jnp.argmax)
    if (l1 > best) { best = l1; s = 1; }
    if (l2 > best) { s = 2; }
    sel[m0 + lane] = s;
  }
}

// ---------------- Kernel 4: fused decompress + residual ---------------------
__global__ __launch_bounds__(256) void dge_out(const float* __restrict__ x,
                                               const float* __restrict__ y,
                                               const float* __restrict__ p2,
                                               const float* __restrict__ p4,
                                               const int* __restrict__ sel,
                                               float* __restrict__ out) {
  int t = blockIdx.x * 256 + threadIdx.x;
  const int total = BB * NN * C4;
  if (t >= total) return;
  int c4 = t % C4;
  int n  = (t / C4) % NN;
  int b  =  t / (C4 * NN);
  int h = n / WW, w = n % WW;
  int hr = h >> 2, wr = w >> 2;

  int s = sel[(b * HR + hr) * WR + wr];

  const float4* x4  = (const float4*)x;
  const float4* y4  = (const float4*)y;
  const float4* p24 = (const float4*)p2;
  const float4* p44 = (const float4*)p4;

  float4 xv = x4[t];
  float4 yv, cv;
  if (s == 0) {
    size_t tok = (size_t)hr * WR + wr;
    yv = y4[((size_t)b * MM + tok) * C4 + c4];
    cv = p44[((size_t)b * HR * WR + tok) * C4 + c4];
  } else if (s == 1) {
    int h2 = h >> 1, w2 = w >> 1;
    size_t tok = (size_t)h2 * W2 + w2;
    yv = y4[((size_t)b * MM + Y_OFF1 + tok) * C4 + c4];
    cv = p24[((size_t)b * H2 * W2 + tok) * C4 + c4];
  } else {
    yv = y4[((size_t)b * MM + Y_OFF2 + n) * C4 + c4];
    cv = xv;
  }
  float4 o;
  o.x = xv.x + (yv.x - cv.x);
  o.y = xv.y + (yv.y - cv.y);
  o.z = xv.z + (yv.z - cv.z);
  o.w = xv.w + (yv.w - cv.w);
  ((float4*)out)[t] = o;
}

// ---------------- Host launcher ---------------------------------------------
extern "C" void kernel_launch(void* const* d_in, const int* in_sizes, int n_in,
                              void* d_out, int out_size, void* d_ws, size_t ws_size,
                              hipStream_t stream) {
  const float* x  = (const float*)d_in[0];
  const float* y  = (const float*)d_in[1];
  const float* gw = (const float*)d_in[2];
  const float* gb = (const float*)d_in[3];
  (void)in_sizes; (void)n_in; (void)out_size; (void)ws_size;

  // Workspace layout
  float* p2  = (float*)d_ws;                                  // 32*784*384 f32
  float* p4  = p2 + (size_t)BB * H2 * W2 * CC;                // 32*196*384 f32
  int*   sel = (int*)(p4 + (size_t)BB * HR * WR * CC);        // 6272 int32

  float* out = (float*)d_out;

  {
    int total = BB * H2 * W2 * C4;                            // 2,408,448
    dge_pool2<<<(total + 255) / 256, 256, 0, stream>>>(x, p2);
  }
  {
    int total = BB * HR * WR * C4;                            // 602,112
    dge_pool4<<<(total + 255) / 256, 256, 0, stream>>>(p2, p4);
  }
  {
    // 392 row-tiles of 16, 8 waves per block -> 49 blocks
    dge_gate<<<49, 256, 0, stream>>>(p4, gw, gb, sel);
  }
  {
    int total = BB * NN * C4;                                 // 9,633,792
    dge_out<<<(total + 255) / 256, 256, 0, stream>>>(x, y, p2, p4, sel, out);
  }
}